// TopCostMatcher_39092792329017
// MI455X (gfx1250) — compile-verified
//
#include <hip/hip_runtime.h>
#include <hip/hip_bf16.h>

typedef __attribute__((ext_vector_type(2))) float v2f;
typedef __attribute__((ext_vector_type(8))) float v8f;

#define ALPHA_F   0.25f
#define BIG_F     1.0e6f
#define NCLS      80
#define KSEL      9

// ---------------------------------------------------------------------------
// Kernel 1: per-inside-pair reductions.
//   dice dot-product via V_WMMA_F32_16X16X4_F32 (diagonal of 16x16 tile),
//   row sums in VALU, polar-IoU term folded in by wave 0.
//   Block = 128 threads = 4 waves; block handles 16 pairs; each wave owns a
//   quarter of the HW*HW axis. Lanes load float4 (b128); each quad feeds two
//   K=4 WMMA chunks: chunk A = positions {8j,8j+1,8j+4,8j+5}, chunk B =
//   {8j+2,8j+3,8j+6,8j+7} (K-ordering is free since we only need the sum).
// ---------------------------------------------------------------------------
__global__ void __launch_bounds__(128) pair_reduce_kernel(
    const float* __restrict__ mask_targ,   // [NIN, HW2]
    const float* __restrict__ mask_pred,   // [NIN, HW2]
    const float* __restrict__ poly_targ,   // [NIN, R]
    const float* __restrict__ poly_pred,   // [NIN, R]
    float* __restrict__ pair_cost,         // [NIN]
    int HW2, int R)
{
    __shared__ float s_red[3][4][16];

    const int wid      = threadIdx.x >> 5;
    const int lane     = threadIdx.x & 31;
    const int rowLane  = lane & 15;          // pair within 16-pair tile
    const int qoff     = (lane >> 4) * 4;    // float offset of this lane's quad
    const int pairBase = blockIdx.x * 16;
    const int segLen   = HW2 >> 2;           // 4 waves split the reduction axis
    const int segBase  = wid * segLen;

    const float4* mp4 = (const float4*)(mask_pred +
        (size_t)(pairBase + rowLane) * HW2 + segBase + qoff);
    const float4* mt4 = (const float4*)(mask_targ +
        (size_t)(pairBase + rowLane) * HW2 + segBase + qoff);

    v8f   acc  = {};       // D[m][n] accumulator; diagonal = per-pair dot
    float accP = 0.0f;     // per-lane partial of sum(mask_pred)
    float accT = 0.0f;     // per-lane partial of sum(mask_targ)

    const int iters = segLen >> 3;           // 8 floats (2 WMMA chunks) / iter
#pragma unroll 4
    for (int j = 0; j < iters; ++j) {
        float4 a4 = mp4[j * 2];              // 16B load
        float4 b4 = mt4[j * 2];
        __builtin_prefetch(((const char*)(mp4 + j * 2)) + 2048, 0, 3);
        __builtin_prefetch(((const char*)(mt4 + j * 2)) + 2048, 0, 3);
        v2f a0; a0.x = a4.x; a0.y = a4.y;
        v2f a1; a1.x = a4.z; a1.y = a4.w;
        v2f b0; b0.x = b4.x; b0.y = b4.y;
        v2f b1; b1.x = b4.z; b1.y = b4.w;
        acc = __builtin_amdgcn_wmma_f32_16x16x4_f32(
                  false, a0, false, b0, (short)0, acc, false, false);
        acc = __builtin_amdgcn_wmma_f32_16x16x4_f32(
                  false, a1, false, b1, (short)0, acc, false, false);
        accP += (a4.x + a4.y) + (a4.z + a4.w);
        accT += (b4.x + b4.y) + (b4.z + b4.w);
    }

    // --- extract diagonal D[m][m] ---------------------------------------
    // C/D layout: VGPR r, lanes 0-15 -> (M=r, N=lane); lanes 16-31 -> (M=8+r, N=lane-16)
    // diag m<8  lives in lane m,    vgpr m
    // diag m>=8 lives in lane m+16, vgpr m-8
    float d0 = acc[0], d1 = acc[1], d2 = acc[2], d3 = acc[3];
    float d4 = acc[4], d5 = acc[5], d6 = acc[6], d7 = acc[7];
    int vidx = (lane < 8) ? lane : ((lane >= 24) ? (lane - 24) : -1);
    float diag = 0.0f;
    diag = (vidx == 0) ? d0 : diag;  diag = (vidx == 1) ? d1 : diag;
    diag = (vidx == 2) ? d2 : diag;  diag = (vidx == 3) ? d3 : diag;
    diag = (vidx == 4) ? d4 : diag;  diag = (vidx == 5) ? d5 : diag;
    diag = (vidx == 6) ? d6 : diag;  diag = (vidx == 7) ? d7 : diag;

    // move diag for pair m to lane m (m < 16)
    int srcLane = (lane < 8) ? lane : (lane + 16);
    float dotm  = __shfl(diag, srcLane, 32);

    // fold row sums across the two quad-halves (lane m + lane m^16 hold row m)
    float spRow = accP + __shfl_xor(accP, 16, 32);
    float stRow = accT + __shfl_xor(accT, 16, 32);

    if (lane < 16) {
        s_red[0][wid][lane] = dotm;
        s_red[1][wid][lane] = spRow;
        s_red[2][wid][lane] = stRow;
    }
    __syncthreads();

    if (wid == 0 && lane < 16) {
        float D = 0.0f, SP = 0.0f, ST = 0.0f;
#pragma unroll
        for (int w = 0; w < 4; ++w) {
            D  += s_red[0][w][lane];
            SP += s_red[1][w][lane];
            ST += s_red[2][w][lane];
        }
        int n = pairBase + lane;
        // polar IoU term
        const float* pp = poly_pred + (size_t)n * R;
        const float* pt = poly_targ + (size_t)n * R;
        float lmax = 0.0f, lmin = 0.0f;
        for (int r = 0; r < R; ++r) {
            float a = pp[r], b = pt[r];
            lmax += fmaxf(a, b);
            lmin += fminf(a, b);
        }
        float dice = (2.0f * D + 1.0f) / (SP + ST + 1.0f);
        pair_cost[n] = __logf(lmax / lmin) + 1.0f - dice;
    }
}

// ---------------------------------------------------------------------------
// Kernel 2: focal classification cost gathered at gt labels, transposed layout
//   Ct[g][p] = focal(p, label_targ[g]) + 2*BIG ; Pm[g][p] = -1
// ---------------------------------------------------------------------------
__global__ void class_cost_kernel(
    const int*   __restrict__ label_targ,  // [G]
    const float* __restrict__ label_pred,  // [P, NCLS]
    float* __restrict__ Ct,                // [G, P]
    int*   __restrict__ Pm,                // [G, P]
    int P, int G)
{
    int p = blockIdx.x * blockDim.x + threadIdx.x;
    if (p >= P) return;
    const float* row = label_pred + (size_t)p * NCLS;
    for (int g = 0; g < G; ++g) {
        int   lt = label_targ[g];
        float x  = row[lt];
        float s  = 1.0f / (1.0f + __expf(-x));
        float neg = (1.0f - ALPHA_F) * s * s * (-__logf(1.0f - s + 1e-8f));
        float pos = ALPHA_F * (1.0f - s) * (1.0f - s) * (-__logf(s + 1e-8f));
        Ct[(size_t)g * P + p] = (pos - neg) + 2.0f * BIG_F;
        Pm[(size_t)g * P + p] = -1;
    }
}

// ---------------------------------------------------------------------------
// Kernel 3: scatter inside-pair costs (pairs are unique by construction)
// ---------------------------------------------------------------------------
__global__ void scatter_kernel(
    const int*   __restrict__ inside_pt,
    const int*   __restrict__ inside_gt,
    const float* __restrict__ pair_cost,
    float* __restrict__ Ct, int* __restrict__ Pm,
    int NIN, int P)
{
    int n = blockIdx.x * blockDim.x + threadIdx.x;
    if (n >= NIN) return;
    int pt = inside_pt[n];
    int gt = inside_gt[n];
    size_t idx = (size_t)gt * P + pt;
    Ct[idx] += pair_cost[n] - 2.0f * BIG_F;   // -> focal + mask + miou
    Pm[idx]  = n;
}

// ---------------------------------------------------------------------------
// Kernel 4: initialize pos_class_targ to background (80)
// ---------------------------------------------------------------------------
__global__ void init_class_kernel(float* __restrict__ out_class, int P)
{
    int p = blockIdx.x * blockDim.x + threadIdx.x;
    if (p < P) out_class[p] = (float)NCLS;
}

// ---------------------------------------------------------------------------
// Kernel 5: top-K smallest per gt column + final outputs
// ---------------------------------------------------------------------------
__global__ void __launch_bounds__(256) topk_kernel(
    const float* __restrict__ Ct,          // [G, P]
    const int*   __restrict__ Pm,          // [G, P]
    const int*   __restrict__ label_targ,  // [G]
    float* __restrict__ out_class,         // [P]
    float* __restrict__ out_pos,           // [KSEL, G]
    float* __restrict__ out_valid,         // [KSEL, G]
    int P, int G)
{
    __shared__ float sv[256 * KSEL];
    __shared__ int   si[256 * KSEL];

    const int g   = blockIdx.x;
    const int tid = threadIdx.x;

    float bv[KSEL];
    int   bi[KSEL];
#pragma unroll
    for (int k = 0; k < KSEL; ++k) { bv[k] = 3.4e38f; bi[k] = -1; }

    const float* col = Ct + (size_t)g * P;
    for (int p = tid; p < P; p += 256) {
        float c = col[p];
        if (c < bv[KSEL - 1]) {
            int j = KSEL - 1;
            while (j > 0 && bv[j - 1] > c) {
                bv[j] = bv[j - 1]; bi[j] = bi[j - 1]; --j;
            }
            bv[j] = c; bi[j] = p;
        }
    }
#pragma unroll
    for (int k = 0; k < KSEL; ++k) {
        sv[tid * KSEL + k] = bv[k];
        si[tid * KSEL + k] = bi[k];
    }
    __syncthreads();

    if (tid == 0) {
        for (int k = 0; k < KSEL; ++k) {
            float best = 3.4e38f; int bidx = -1, slot = -1;
            for (int t = 0; t < 256 * KSEL; ++t) {
                if (sv[t] < best) { best = sv[t]; bidx = si[t]; slot = t; }
            }
            if (slot >= 0) sv[slot] = 3.4e38f;   // remove winner
            bool valid = (best < BIG_F) && (bidx >= 0);
            out_valid[k * G + g] = valid ? 1.0f : 0.0f;
            out_pos[k * G + g]   = valid ? (float)Pm[(size_t)g * P + bidx] : -1.0f;
            if (valid) out_class[bidx] = (float)label_targ[g];
        }
    }
}

// ---------------------------------------------------------------------------
extern "C" void kernel_launch(void* const* d_in, const int* in_sizes, int n_in,
                              void* d_out, int out_size, void* d_ws, size_t ws_size,
                              hipStream_t stream)
{
    const int*   label_targ = (const int*)  d_in[0];
    const float* label_pred = (const float*)d_in[1];
    const float* poly_targ  = (const float*)d_in[2];
    const float* poly_pred  = (const float*)d_in[3];
    const float* mask_targ  = (const float*)d_in[4];
    const float* mask_pred  = (const float*)d_in[5];
    const int*   inside_pt  = (const int*)  d_in[6];
    const int*   inside_gt  = (const int*)  d_in[7];

    const int G   = in_sizes[0];
    const int P   = in_sizes[1] / NCLS;
    const int NIN = in_sizes[6];
    const int R   = in_sizes[2] / NIN;
    const int HW2 = in_sizes[4] / NIN;        // 4096

    // workspace layout (floats/ints)
    float* pair_cost = (float*)d_ws;                       // [NIN]
    float* Ct        = pair_cost + NIN;                    // [G, P]
    int*   Pm        = (int*)(Ct + (size_t)G * P);         // [G, P]

    float* out       = (float*)d_out;
    float* out_class = out;                                // [P]
    float* out_pos   = out + P;                            // [KSEL, G]
    float* out_valid = out + P + KSEL * G;                 // [KSEL, G]

    // 1) per-pair dice (WMMA) + polar IoU
    pair_reduce_kernel<<<NIN / 16, 128, 0, stream>>>(
        mask_targ, mask_pred, poly_targ, poly_pred, pair_cost, HW2, R);

    // 2) focal class cost -> transposed cost matrix
    class_cost_kernel<<<(P + 255) / 256, 256, 0, stream>>>(
        label_targ, label_pred, Ct, Pm, P, G);

    // 3) scatter inside pairs
    scatter_kernel<<<(NIN + 255) / 256, 256, 0, stream>>>(
        inside_pt, inside_gt, pair_cost, Ct, Pm, NIN, P);

    // 4) init class targets to background
    init_class_kernel<<<(P + 255) / 256, 256, 0, stream>>>(out_class, P);

    // 5) top-K per gt + outputs
    topk_kernel<<<G, 256, 0, stream>>>(
        Ct, Pm, label_targ, out_class, out_pos, out_valid, P, G);
}